// FuseCausal_81965155877006
// MI455X (gfx1250) — compile-verified
//
#include <hip/hip_runtime.h>
#include <hip/hip_bf16.h>
#include <math.h>

#define N_ROWS 2048
#define D_DIM  1024
#define C_CLS  10
#define GCE_Q  0.7f
#define EPS_W  1e-8f

typedef float v2f __attribute__((ext_vector_type(2)));
typedef float v8f __attribute__((ext_vector_type(8)));

// ---------------------------------------------------------------------------
// Kernel 1: projection GEMM via f32 WMMA 16x16x4.
// One wave per 16-row tile. blockIdx.y selects (Z, W) combo:
//   0: z_c @ Wc.T  -> Lc     1: z_b @ Wc.T -> Rc
//   2: z_c @ Wb.T  -> Lb     3: z_b @ Wb.T -> Rb
// Output padded to 16 cols (cols 10..15 are exact zeros since B rows >= C are
// zeroed). Workspace layout: 4 matrices of N_ROWS*16 floats.
// ---------------------------------------------------------------------------
__global__ __launch_bounds__(32) void proj_wmma(
    const float* __restrict__ z_c, const float* __restrict__ z_b,
    const float* __restrict__ Wc,  const float* __restrict__ Wb,
    float* __restrict__ ws)
{
    const int combo = blockIdx.y;                       // 0..3
    const float* __restrict__ Z = (combo & 1) ? z_b : z_c;
    const float* __restrict__ W = (combo >> 1) ? Wb : Wc;
    float* __restrict__ out = ws + (size_t)combo * (N_ROWS * 16);

    const int lane  = threadIdx.x;      // 0..31
    const int m     = lane & 15;        // A row / B col within tile
    const int khalf = (lane >> 4) << 1; // 0 for lanes 0-15, 2 for lanes 16-31
    const int row   = blockIdx.x * 16 + m;

    const float* __restrict__ zrow = Z + (size_t)row * D_DIM + khalf;
    // clamp pointer for n >= C (avoid OOB), zero the fragment instead
    const int wvalid = (m < C_CLS);
    const float* __restrict__ wrow =
        W + (size_t)(wvalid ? m : 0) * D_DIM + khalf;

    v8f acc = {};
#pragma unroll 4
    for (int k = 0; k < D_DIM; k += 4) {
        // A fragment: z[row][k+khalf], z[row][k+khalf+1]
        v2f a = *(const v2f*)(zrow + k);
        // B fragment: B[kk][n] = W[n][kk]  ->  W[m][k+khalf+{0,1}]
        v2f bw = *(const v2f*)(wrow + k);
        v2f b;
        b.x = wvalid ? bw.x : 0.0f;
        b.y = wvalid ? bw.y : 0.0f;
        acc = __builtin_amdgcn_wmma_f32_16x16x4_f32(
            /*neg_a=*/false, a, /*neg_b=*/false, b,
            /*c_mod=*/(short)0, acc, /*reuse_a=*/false, /*reuse_b=*/false);
    }

    // C/D layout: VGPR v -> row (v + 8*(lane>>4)), col (lane&15)
    const int colN = lane & 15;
    const int rbase = blockIdx.x * 16 + ((lane >> 4) << 3);
#pragma unroll
    for (int v = 0; v < 8; ++v) {
        out[(size_t)(rbase + v) * 16 + colN] = acc[v];
    }
}

// ---------------------------------------------------------------------------
// Kernel 2: N x N pairwise fused loss from staged projections.
// Block = 16x16 threads covering a 16x16 (i,j) tile.
// ---------------------------------------------------------------------------
__global__ __launch_bounds__(256) void pair_loss(
    const float* __restrict__ Lc, const float* __restrict__ Rc,
    const float* __restrict__ Lb, const float* __restrict__ Rb,
    const float* __restrict__ bc, const float* __restrict__ bb,
    const int*   __restrict__ y,
    float* __restrict__ out)
{
    __shared__ float sLc[16][16];
    __shared__ float sRc[16][16];
    __shared__ float sLb[16][16];
    __shared__ float sRb[16][16];
    __shared__ float sbc[16];
    __shared__ float sbb[16];
    __shared__ int   syi[16];
    __shared__ int   syj[16];

    const int jtile = blockIdx.x;
    const int itile = blockIdx.y;
    const int c = threadIdx.x;          // 0..15
    const int r = threadIdx.y;          // 0..15
    const int t = r * 16 + c;

    const int gi = itile * 16 + r;
    const int gj = jtile * 16 + r;
    sLc[r][c] = Lc[(size_t)gi * 16 + c];
    sLb[r][c] = Lb[(size_t)gi * 16 + c];
    sRc[r][c] = Rc[(size_t)gj * 16 + c];
    sRb[r][c] = Rb[(size_t)gj * 16 + c];
    if (t < C_CLS)              { sbc[t] = bc[t]; sbb[t] = bb[t]; }
    if (t >= 16 && t < 32)      syi[t - 16] = y[itile * 16 + (t - 16)];
    if (t >= 32 && t < 48)      syj[t - 32] = y[jtile * 16 + (t - 32)];
    __syncthreads();

    const int i = r;                    // pair row within tile
    const int j = c;                    // pair col within tile
    const int yi = syi[i];
    const int yj = syj[j];

    // pass 1: logits, maxes, and label-selected logits (select, no dyn index)
    float lcv[C_CLS], lbv[C_CLS];
    float mc = -INFINITY, mb = -INFINITY;
    float lc_yi = 0.0f, lb_yi = 0.0f, lb_yj = 0.0f;
#pragma unroll
    for (int k = 0; k < C_CLS; ++k) {
        const float lc = sLc[i][k] + sRc[j][k] + sbc[k];
        const float lb = sLb[i][k] + sRb[j][k] + sbb[k];
        lcv[k] = lc;
        lbv[k] = lb;
        mc = fmaxf(mc, lc);
        mb = fmaxf(mb, lb);
        lc_yi = (k == yi) ? lc : lc_yi;
        lb_yi = (k == yi) ? lb : lb_yi;
        lb_yj = (k == yj) ? lb : lb_yj;
    }

    // pass 2: stable log-sum-exp
    float sc = 0.0f, sb = 0.0f;
#pragma unroll
    for (int k = 0; k < C_CLS; ++k) {
        sc += __expf(lcv[k] - mc);
        sb += __expf(lbv[k] - mb);
    }
    const float lse_c = __logf(sc) + mc;
    const float lse_b = __logf(sb) + mb;

    const float ce_pc   = lse_c - lc_yi;
    const float ce_pb   = lse_b - lb_yi;
    const float logp_yj = lb_yj - lse_b;
    // GCE = (-logp) * (p ** q) = (-logp) * exp(q * logp)
    const float gce_pb  = (-logp_yj) * __expf(GCE_Q * logp_yj);

    const float w = ce_pb / (ce_pc + ce_pb + EPS_W);
    out[(size_t)(itile * 16 + i) * N_ROWS + (jtile * 16 + j)] =
        fmaf(w, ce_pc, (1.0f - w) * gce_pb);
}

// ---------------------------------------------------------------------------
extern "C" void kernel_launch(void* const* d_in, const int* in_sizes, int n_in,
                              void* d_out, int out_size, void* d_ws, size_t ws_size,
                              hipStream_t stream) {
    const float* z_c = (const float*)d_in[0];
    const float* z_b = (const float*)d_in[1];
    const float* Wc  = (const float*)d_in[2];
    const float* bc  = (const float*)d_in[3];
    const float* Wb  = (const float*)d_in[4];
    const float* bb  = (const float*)d_in[5];
    const int*   y   = (const int*)d_in[6];
    float* out = (float*)d_out;

    float* ws = (float*)d_ws;
    float* Lc = ws + 0 * (N_ROWS * 16);
    float* Rc = ws + 1 * (N_ROWS * 16);
    float* Lb = ws + 2 * (N_ROWS * 16);
    float* Rb = ws + 3 * (N_ROWS * 16);

    // 1) four projection GEMMs via f32 WMMA: grid (128 row tiles, 4 combos)
    dim3 g1(N_ROWS / 16, 4);
    proj_wmma<<<g1, 32, 0, stream>>>(z_c, z_b, Wc, Wb, ws);

    // 2) pairwise fused loss over the 2048x2048 grid of (i, j) pairs
    dim3 g2(N_ROWS / 16, N_ROWS / 16);
    dim3 b2(16, 16);
    pair_loss<<<g2, b2, 0, stream>>>(Lc, Rc, Lb, Rb, bc, bb, y, out);
}